// ImplicitRaymarchingIntegrandSlang_5222680232014
// MI455X (gfx1250) — compile-verified
//
#include <hip/hip_runtime.h>
#include <math.h>

#define GRES   64
#define FOVF   0.6f
#define STEPF  (2.0f / 500.0f)
#define NSTEPS 500
#define T0F    0.5f
#define EPSF   1e-8f

typedef __attribute__((ext_vector_type(2))) float v2f;
typedef __attribute__((ext_vector_type(8))) float v8f;

// _sdf: trilinear sample of the 64^3 grid on [-1,1]^3; returns 1.0 outside [-1,1]^3.
__device__ __forceinline__ float sdf_eval(const float* __restrict__ g,
                                          float px, float py, float pz) {
    if (!(fabsf(px) <= 1.0f && fabsf(py) <= 1.0f && fabsf(pz) <= 1.0f))
        return 1.0f;
    float gx = fminf(fmaxf((px + 1.0f) * 0.5f, 0.0f), 1.0f) * (float)(GRES - 1);
    float gy = fminf(fmaxf((py + 1.0f) * 0.5f, 0.0f), 1.0f) * (float)(GRES - 1);
    float gz = fminf(fmaxf((pz + 1.0f) * 0.5f, 0.0f), 1.0f) * (float)(GRES - 1);
    int ix = (int)gx; if (ix > GRES - 2) ix = GRES - 2;
    int iy = (int)gy; if (iy > GRES - 2) iy = GRES - 2;
    int iz = (int)gz; if (iz > GRES - 2) iz = GRES - 2;
    float fx = gx - (float)ix, fy = gy - (float)iy, fz = gz - (float)iz;
    const float* __restrict__ p0 = g + (((ix << 6) + iy) << 6) + iz;
    float v000 = p0[0],            v001 = p0[1];
    float v010 = p0[64],           v011 = p0[65];
    float v100 = p0[4096],         v101 = p0[4097];
    float v110 = p0[4160],         v111 = p0[4161];
    float c00 = v000 + fz * (v001 - v000);
    float c01 = v010 + fz * (v011 - v010);
    float c10 = v100 + fz * (v101 - v100);
    float c11 = v110 + fz * (v111 - v110);
    float c0 = c00 + fy * (c01 - c00);
    float c1 = c10 + fy * (c11 - c10);
    return c0 + fx * (c1 - c0);
}

__global__ __launch_bounds__(256)
void raymarch_kernel(const float* __restrict__ xin,
                     const float* __restrict__ grid,
                     const float* __restrict__ cam3,
                     const float* __restrict__ ldir3,
                     const float* __restrict__ lcol3,
                     const float* __restrict__ dcol3,
                     float* __restrict__ out, int n) {
    const int ray  = blockIdx.x * blockDim.x + threadIdx.x;
    const int lane = threadIdx.x & 31;

    __builtin_prefetch(grid, 0, 3);   // global_prefetch_b8: pull grid toward caches

    // ---- camera basis (uniform) ----
    const float ocx = cam3[0], ocy = cam3[1], ocz = cam3[2];
    float cn = sqrtf(ocx * ocx + ocy * ocy + ocz * ocz) + EPSF;
    float fwx = -ocx / cn, fwy = -ocy / cn, fwz = -ocz / cn;
    // right = normalize(cross(fwd, up=(0,0,1))) = normalize(fwy, -fwx, 0)
    float rx = fwy, ry = -fwx;
    float rn = sqrtf(rx * rx + ry * ry) + EPSF;
    rx /= rn; ry /= rn;
    const float rz = 0.0f;
    // upv = cross(right, fwd)
    float ux = ry * fwz - rz * fwy;
    float uy = rz * fwx - rx * fwz;
    float uz = rx * fwy - ry * fwx;

    // ---- per-ray NDC ----
    const int ri = (ray < n) ? ray : (n - 1);
    const float ndcx = (xin[2 * ri + 0] - 0.5f) * 2.0f;
    const float ndcy = (xin[2 * ri + 1] - 0.5f) * 2.0f;

    // ---- ray-direction generation via V_WMMA_F32_16X16X4_F32 ----
    // D[m,n] = FOV*right[m]*ndcx[n] + FOV*upv[m]*ndcy[n] + fwd[m]
    // A (16x4): rows 0..2 = [FOV*right | FOV*upv | fwd | 0], rows 3..15 = 0
    // B (4x16): [ndcx; ndcy; 1; 0] per ray column.
    const int  m  = lane & 15;
    const bool hi = lane >= 16;      // lanes 16..31 hold K=2,3 halves
    float rc = (m == 0) ? rx : (m == 1) ? ry : rz;
    float uc = (m == 0) ? ux : (m == 1) ? uy : uz;
    float fc = (m == 0) ? fwx : (m == 1) ? fwy : fwz;
    const bool rowOK = (m < 3);
    v2f a;
    a.x = rowOK ? (hi ? fc : FOVF * rc) : 0.0f;   // K=0 (lo) / K=2 (hi)
    a.y = rowOK ? (hi ? 0.0f : FOVF * uc) : 0.0f; // K=1 (lo) / K=3 (hi)

    float nx_lo = __shfl(ndcx, m, 32);
    float ny_lo = __shfl(ndcy, m, 32);
    float nx_hi = __shfl(ndcx, 16 + m, 32);
    float ny_hi = __shfl(ndcy, 16 + m, 32);
    v2f b0, b1;
    b0.x = hi ? 1.0f : nx_lo;  b0.y = hi ? 0.0f : ny_lo;   // rays 0..15 of wave
    b1.x = hi ? 1.0f : nx_hi;  b1.y = hi ? 0.0f : ny_hi;   // rays 16..31 of wave
    v8f zc = {};
    v8f dm0 = __builtin_amdgcn_wmma_f32_16x16x4_f32(false, a, false, b0,
                                                    (short)0, zc, false, false);
    v8f dm1 = __builtin_amdgcn_wmma_f32_16x16x4_f32(false, a, false, b1,
                                                    (short)0, zc, false, false);
    // lanes 0..15: ray = lane, components in dm0[0..2]
    // lanes 16..31: ray's components live in lane (lane-16) of dm1[0..2]
    float d1x = __shfl(dm1[0], m, 32);
    float d1y = __shfl(dm1[1], m, 32);
    float d1z = __shfl(dm1[2], m, 32);
    float dx = hi ? d1x : dm0[0];
    float dy = hi ? d1y : dm0[1];
    float dz = hi ? d1z : dm0[2];
    float dn = sqrtf(dx * dx + dy * dy + dz * dz) + EPSF;
    dx /= dn; dy /= dn; dz /= dn;

    // ---- analytic ray/[-1,1]^3 slab intersection (cull + step-range clamp) ----
    float idx_ = 1.0f / dx, idy_ = 1.0f / dy, idz_ = 1.0f / dz;
    float tx1 = (-1.0f - ocx) * idx_, tx2 = (1.0f - ocx) * idx_;
    float ty1 = (-1.0f - ocy) * idy_, ty2 = (1.0f - ocy) * idy_;
    float tz1 = (-1.0f - ocz) * idz_, tz2 = (1.0f - ocz) * idz_;
    float tmin = fmaxf(fmaxf(fminf(tx1, tx2), fminf(ty1, ty2)), fminf(tz1, tz2));
    float tmax = fminf(fminf(fmaxf(tx1, tx2), fmaxf(ty1, ty2)), fmaxf(tz1, tz2));

    int kstart = 0, kend = -1;
    if (tmax >= tmin && tmax >= T0F) {
        kstart = (int)floorf((tmin - T0F) / STEPF) - 1;
        if (kstart < 0) kstart = 0;
        kend = (int)ceilf((tmax - T0F) / STEPF) + 1;
        if (kend > NSTEPS - 1) kend = NSTEPS - 1;
    }

    // ---- coarse march: latch first (prev>0 && s<=0) crossing, break early ----
    bool hit = false;
    int khit = 0;
    float prev = 1.0e3f;
    #pragma unroll 1
    for (int k = kstart; k <= kend; ++k) {
        float t = fmaf((float)k, STEPF, T0F);
        float s = sdf_eval(grid, fmaf(dx, t, ocx), fmaf(dy, t, ocy), fmaf(dz, t, ocz));
        if (prev > 0.0f && s <= 0.0f) { hit = true; khit = k; break; }
        prev = s;
    }

    float r0 = 0.0f, r1 = 0.0f, r2 = 0.0f;
    if (hit) {
        // ---- secant refinement between bracketing samples ----
        float t1 = fmaf((float)khit, STEPF, T0F);
        float t0 = t1 - STEPF;
        float s0 = sdf_eval(grid, fmaf(dx, t0, ocx), fmaf(dy, t0, ocy), fmaf(dz, t0, ocz));
        float s1 = sdf_eval(grid, fmaf(dx, t1, ocx), fmaf(dy, t1, ocy), fmaf(dz, t1, ocz));
        float alpha = fminf(fmaxf(s0 / (s0 - s1 + EPSF), 0.0f), 1.0f);
        float tr = t0 + STEPF * alpha;
        float px = fmaf(dx, tr, ocx), py = fmaf(dy, tr, ocy), pz = fmaf(dz, tr, ocz);

        // ---- normal: central differences, h = one voxel in world units ----
        const float h = 2.0f / (float)GRES;
        float gx = sdf_eval(grid, px + h, py, pz) - sdf_eval(grid, px - h, py, pz);
        float gy = sdf_eval(grid, px, py + h, pz) - sdf_eval(grid, px, py - h, pz);
        float gz = sdf_eval(grid, px, py, pz + h) - sdf_eval(grid, px, py, pz - h);
        float gn = sqrtf(gx * gx + gy * gy + gz * gz) + EPSF;
        gx /= gn; gy /= gn; gz /= gn;

        // ---- Lambert ----
        float lx = ldir3[0], ly = ldir3[1], lz = ldir3[2];
        float ln = sqrtf(lx * lx + ly * ly + lz * lz) + EPSF;
        lx = -lx / ln; ly = -ly / ln; lz = -lz / ln;
        float lam = fmaxf(gx * lx + gy * ly + gz * lz, 0.0f);
        r0 = dcol3[0] * lcol3[0] * lam;
        r1 = dcol3[1] * lcol3[1] * lam;
        r2 = dcol3[2] * lcol3[2] * lam;
    }

    if (ray < n) {
        out[3 * ray + 0] = r0;
        out[3 * ray + 1] = r1;
        out[3 * ray + 2] = r2;
    }
}

extern "C" void kernel_launch(void* const* d_in, const int* in_sizes, int n_in,
                              void* d_out, int out_size, void* d_ws, size_t ws_size,
                              hipStream_t stream) {
    (void)n_in; (void)d_ws; (void)ws_size; (void)out_size;
    const float* x    = (const float*)d_in[0];   // [N,2]
    const float* grid = (const float*)d_in[1];   // [64,64,64]
    const float* cam  = (const float*)d_in[2];   // [3]
    const float* ldir = (const float*)d_in[3];   // [3]
    const float* lcol = (const float*)d_in[4];   // [3]
    const float* dcol = (const float*)d_in[5];   // [3]
    float* out = (float*)d_out;                  // [N,3]
    const int n = in_sizes[0] / 2;
    const int threads = 256;
    const int blocks = (n + threads - 1) / threads;
    raymarch_kernel<<<blocks, threads, 0, stream>>>(x, grid, cam, ldir, lcol, dcol, out, n);
}